// SSM_70489003262404
// MI455X (gfx1250) — compile-verified
//
#include <hip/hip_runtime.h>
#include <cstdint>
#include <cstddef>

// Problem dims (match reference)
#define V_DIM 32
#define T_DIM 8192
#define L_DIM 16
#define JITTER_F 1e-6f
#define LOG2PI_F 1.8378770664093453f
#define SQRT3_F 1.7320508075688772f

#define TBF 32 // forward time-tile staged in LDS
#define TBB 16 // backward time-tile staged in LDS

// ---------------- CDNA5 async global->LDS path ----------------
#if defined(__gfx1250__) && __has_builtin(__builtin_amdgcn_global_load_async_to_lds_b128)
#define HAVE_ASYNC_LDS 1
#else
#define HAVE_ASYNC_LDS 0
#endif

typedef int v4i __attribute__((ext_vector_type(4)));
typedef __attribute__((address_space(1))) v4i* gptr_v4i;
typedef __attribute__((address_space(3))) v4i* lptr_v4i;

__device__ __forceinline__ void async_copy_b128(const float* gsrc, float* lds_dst) {
#if HAVE_ASYNC_LDS
  const __attribute__((address_space(1))) float* g1 =
      (const __attribute__((address_space(1))) float*)gsrc;
  __attribute__((address_space(3))) float* l3 =
      (__attribute__((address_space(3))) float*)lds_dst;
  __builtin_amdgcn_global_load_async_to_lds_b128(
      (gptr_v4i)g1, (lptr_v4i)l3, /*imm offset*/ 0, /*cpol*/ 0);
#else
  *(float4*)lds_dst = *(const float4*)gsrc;
#endif
}

// Wait until <= N async ops outstanding. Both kernels put ONLY loads on
// ASYNCcnt, and async loads retire in order -> threshold waits are exact.
template <int N>
__device__ __forceinline__ void wait_async_le() {
#if HAVE_ASYNC_LDS
#if __has_builtin(__builtin_amdgcn_s_wait_asynccnt)
  __builtin_amdgcn_s_wait_asynccnt(N);
#else
  asm volatile("s_wait_asynccnt %0" ::"i"(N) : "memory");
#endif
#endif
}

// ---------------- Forward Kalman filter ----------------
// One chain (v,l) per lane. Block = 32 lanes = 2 v's x 16 l's.
// Inputs y/r have layout (V,T,L): per v the TBF-step tile is one contiguous
// span of TBF*16 floats -> double-buffered async LDS copies (16 ops/lane/tile).
__global__ __launch_bounds__(32) void ssm_fwd_kernel(
    const float* __restrict__ dt,    // (V, T-1)
    const float* __restrict__ ypy,   // (V, T, L)
    const float* __restrict__ pvar,  // (V, T, L)
    const float* __restrict__ ls,    // (L)
    const float* __restrict__ var,   // (L)
    float* __restrict__ m_fs,        // (V, L, T, 2)
    float* __restrict__ P_fs,        // (V, L, T, 2, 2)
    float* __restrict__ sum_log_p)   // (V)
{
  __shared__ float ytile[2][2][TBF][L_DIM];
  __shared__ float rtile[2][2][TBF][L_DIM];

  const int lane  = threadIdx.x;
  const int c     = blockIdx.x * 32 + lane; // chain id = v*16 + l
  const int v     = c >> 4;
  const int l     = c & 15;
  const int vslot = lane >> 4;
  const int v0    = blockIdx.x * 2;

  const float lam = SQRT3_F / ls[l];
  const float p0a = var[l];
  const float p0b = p0a * lam * lam;

  auto copy_tile = [&](int buf, int t0) {
    for (int k = lane; k < 2 * TBF * L_DIM / 4; k += 32) { // 256 chunks
      const int vs  = k >> 7;               // 128 chunks per v-region
      const int rem = (k & 127) << 2;       // float offset within region
      const size_t goff = ((size_t)(v0 + vs) * T_DIM + (size_t)t0) * L_DIM + rem;
      async_copy_b128(ypy  + goff, &ytile[buf][vs][0][0] + rem);
      async_copy_b128(pvar + goff, &rtile[buf][vs][0][0] + rem);
    }
  };

  copy_tile(0, 0);

  // carry: m = 0, P = P0 = diag(var, var*lam^2)
  float m0 = 0.f, m1 = 0.f;
  float P00 = p0a, P01 = 0.f, P11 = p0b;
  float logp = 0.f;

  const size_t mbase = (size_t)c * T_DIM * 2;
  const size_t pbase = (size_t)c * T_DIM * 4;
  const size_t dtbase = (size_t)v * (T_DIM - 1);

  int buf = 0;
  for (int t0 = 0; t0 < T_DIM; t0 += TBF) {
    const bool more = (t0 + TBF) < T_DIM;
    if (more) {
      copy_tile(buf ^ 1, t0 + TBF); // 16 ops/lane in flight for next tile
      wait_async_le<16>();          // current tile complete
    } else {
      wait_async_le<0>();
    }

    for (int tt = 0; tt < TBF; ++tt) {
      const int t = t0 + tt;
      // dt_f = [0, dt_nonzero]
      const float dtv = (t == 0) ? 0.f : dt[dtbase + (t - 1)];
      // Matern-3/2 transition A(dt) and process noise Q = P0 - A P0 A^T
      const float ldt = lam * dtv;
      const float e   = __expf(-ldt);
      const float A00 = e * (ldt + 1.f);
      const float A01 = e * dtv;
      const float A10 = -e * lam * ldt;
      const float A11 = e * (1.f - ldt);
      const float Q00 = p0a - (A00 * A00 * p0a + A01 * A01 * p0b);
      const float Q01 = -(A00 * A10 * p0a + A01 * A11 * p0b);
      const float Q11 = p0b - (A10 * A10 * p0a + A11 * A11 * p0b);
      // predict
      const float mp0 = A00 * m0 + A01 * m1;
      const float mp1 = A10 * m0 + A11 * m1;
      const float T00 = A00 * P00 + A01 * P01;
      const float T01 = A00 * P01 + A01 * P11;
      const float T10 = A10 * P00 + A11 * P01;
      const float T11 = A10 * P01 + A11 * P11;
      const float Pp00 = T00 * A00 + T01 * A01 + Q00;
      const float Pp01 = T00 * A10 + T01 * A11 + Q01;
      const float Pp11 = T10 * A10 + T11 * A11 + Q11;
      // update with observation (H = [1 0])
      const float yv = ytile[buf][vslot][tt][l];
      const float rv = rtile[buf][vslot][tt][l];
      const float s  = Pp00 + rv;
      const float d  = yv - mp0;
      logp += -0.5f * (d * d / s + __logf(s) + LOG2PI_F);
      const float si = 1.f / (s + JITTER_F);
      const float W0 = Pp00 * si;
      const float W1 = Pp01 * si;
      m0 = mp0 + W0 * d;
      m1 = mp1 + W1 * d;
      P00 = Pp00 - W0 * Pp00;
      P01 = Pp01 - W0 * Pp01;
      P11 = Pp11 - W1 * Pp01;
      // store filtered state (chain-contiguous layout (V,L,T,...))
      *(float2*)(m_fs + mbase + (size_t)t * 2) = make_float2(m0, m1);
      *(float4*)(P_fs + pbase + (size_t)t * 4) = make_float4(P00, P01, P01, P11);
    }
    buf ^= 1;
  }

  // sum_log_p[v] = sum over t (done) and over the 16 lanes sharing v
  logp += __shfl_down(logp, 8, 16);
  logp += __shfl_down(logp, 4, 16);
  logp += __shfl_down(logp, 2, 16);
  logp += __shfl_down(logp, 1, 16);
  if (l == 0) sum_log_p[v] = logp;
}

// ---------------- Backward RTS smoother ----------------
// Re-reads m_fs/P_fs (~100MB, L2-resident). Each chain's TBB-step tile region
// is contiguous -> cooperative coalesced async b128 copies into LDS,
// double-buffered (24 ops/lane/tile, <=48 outstanding < 63 counter cap).
#define MROW (TBB * 2 + 4) // padded LDS row strides (16B aligned, de-conflicted)
#define PROW (TBB * 4 + 4)

__global__ __launch_bounds__(32) void ssm_bwd_kernel(
    const float* __restrict__ dt,    // (V, T-1)
    const float* __restrict__ ls,    // (L)
    const float* __restrict__ var,   // (L)
    const float* __restrict__ m_fs,  // (V, L, T, 2)
    const float* __restrict__ P_fs,  // (V, L, T, 2, 2)
    float* __restrict__ mZ,          // (V, L, T)
    float* __restrict__ PZ)          // (V, L, T)
{
  __shared__ float mtile[2][32 * MROW];
  __shared__ float Ptile[2][32 * PROW];

  const int lane  = threadIdx.x;
  const int cbase = blockIdx.x * 32;
  const int c     = cbase + lane;
  const int v     = c >> 4;
  const int l     = c & 15;

  const float lam = SQRT3_F / ls[l];
  const float p0a = var[l];
  const float p0b = p0a * lam * lam;

  const size_t zbase  = (size_t)c * T_DIM;
  const size_t dtbase = (size_t)v * (T_DIM - 1);

  // Cooperative tile copy: m = 32 chains x 8 chunks, P = 32 chains x 16 chunks
  auto copy_tile = [&](int buf, int t0) {
    for (int k = lane; k < 32 * (TBB * 2 / 4); k += 32) {   // 256 chunks
      const int ch = k >> 3;
      const int q  = (k & 7) << 2;
      const size_t goff = ((size_t)(cbase + ch) * T_DIM + (size_t)t0) * 2 + q;
      async_copy_b128(m_fs + goff, &mtile[buf][ch * MROW] + q);
    }
    for (int k = lane; k < 32 * (TBB * 4 / 4); k += 32) {   // 512 chunks
      const int ch = k >> 4;
      const int q  = (k & 15) << 2;
      const size_t goff = ((size_t)(cbase + ch) * T_DIM + (size_t)t0) * 4 + q;
      async_copy_b128(P_fs + goff, &Ptile[buf][ch * PROW] + q);
    }
  };

  // carry0 = filtered state at t = T-1 (direct read)
  const float2 mlast = *(const float2*)(m_fs + ((size_t)c * T_DIM + (T_DIM - 1)) * 2);
  const float4 Plast = *(const float4*)(P_fs + ((size_t)c * T_DIM + (T_DIM - 1)) * 4);
  float ms0 = mlast.x, ms1 = mlast.y;
  float Ps00 = Plast.x, Ps01 = Plast.y, Ps11 = Plast.w;

  copy_tile(0, T_DIM - TBB);

  int buf = 0;
  for (int t0 = T_DIM - TBB; t0 >= 0; t0 -= TBB) {
    const bool more = t0 > 0;
    if (more) {
      copy_tile(buf ^ 1, t0 - TBB); // 24 ops/lane in flight for older tile
      wait_async_le<24>();          // current tile complete
    } else {
      wait_async_le<0>();
    }

    const float* mrow = &mtile[buf][lane * MROW];
    const float* Prow = &Ptile[buf][lane * PROW];

    for (int tt = TBB - 1; tt >= 0; --tt) {
      const int t = t0 + tt;
      const float mf0 = mrow[tt * 2 + 0];
      const float mf1 = mrow[tt * 2 + 1];
      const float Pf00 = Prow[tt * 4 + 0];
      const float Pf01 = Prow[tt * 4 + 1];
      const float Pf11 = Prow[tt * 4 + 3];

      // dt_s = [dt_nonzero, 0]
      const float dtv = (t == T_DIM - 1) ? 0.f : dt[dtbase + t];
      const float ldt = lam * dtv;
      const float e   = __expf(-ldt);
      const float A00 = e * (ldt + 1.f);
      const float A01 = e * dtv;
      const float A10 = -e * lam * ldt;
      const float A11 = e * (1.f - ldt);
      const float Q00 = p0a - (A00 * A00 * p0a + A01 * A01 * p0b);
      const float Q01 = -(A00 * A10 * p0a + A01 * A11 * p0b);
      const float Q11 = p0b - (A10 * A10 * p0a + A11 * A11 * p0b);

      const float mp0 = A00 * mf0 + A01 * mf1;
      const float mp1 = A10 * mf0 + A11 * mf1;
      // B = A @ P_f
      const float B00 = A00 * Pf00 + A01 * Pf01;
      const float B01 = A00 * Pf01 + A01 * Pf11;
      const float B10 = A10 * Pf00 + A11 * Pf01;
      const float B11 = A10 * Pf01 + A11 * Pf11;
      // P_p = B A^T + Q
      const float Pp00 = B00 * A00 + B01 * A01 + Q00;
      const float Pp01 = B00 * A10 + B01 * A11 + Q01;
      const float Pp11 = B10 * A10 + B11 * A11 + Q11;
      // G = (solve(P_p + J I, B))^T via explicit 2x2 inverse
      const float M00 = Pp00 + JITTER_F;
      const float M01 = Pp01;
      const float M11 = Pp11 + JITTER_F;
      const float inv_det = 1.f / (M00 * M11 - M01 * M01);
      const float X00 = (M11 * B00 - M01 * B10) * inv_det;
      const float X01 = (M11 * B01 - M01 * B11) * inv_det;
      const float X10 = (M00 * B10 - M01 * B00) * inv_det;
      const float X11 = (M00 * B11 - M01 * B01) * inv_det;
      const float G00 = X00, G01 = X10, G10 = X01, G11 = X11;
      // m_s' = m_f + G (m_s - m_p)
      const float dm0 = ms0 - mp0, dm1 = ms1 - mp1;
      const float ns0 = mf0 + G00 * dm0 + G01 * dm1;
      const float ns1 = mf1 + G10 * dm0 + G11 * dm1;
      // P_s' = P_f - G (P_s - P_p) G^T
      const float D00 = Ps00 - Pp00, D01 = Ps01 - Pp01, D11 = Ps11 - Pp11;
      const float E00 = G00 * D00 + G01 * D01;
      const float E01 = G00 * D01 + G01 * D11;
      const float E10 = G10 * D00 + G11 * D01;
      const float E11 = G10 * D01 + G11 * D11;
      const float F00 = E00 * G00 + E01 * G01;
      const float F01 = E00 * G10 + E01 * G11;
      const float F11 = E10 * G10 + E11 * G11;
      ms0 = ns0; ms1 = ns1;
      Ps00 = Pf00 - F00;
      Ps01 = Pf01 - F01;
      Ps11 = Pf11 - F11;

      mZ[zbase + t] = ms0;   // H m_s'
      PZ[zbase + t] = Ps00;  // H P_s' H^T
    }
    buf ^= 1;
  }
}

// ---------------- launch ----------------
extern "C" void kernel_launch(void* const* d_in, const int* in_sizes, int n_in,
                              void* d_out, int out_size, void* d_ws, size_t ws_size,
                              hipStream_t stream) {
  (void)in_sizes; (void)n_in; (void)out_size; (void)d_ws; (void)ws_size;
  const float* dt   = (const float*)d_in[0]; // (V, T-1, 1,1,1)
  const float* ypy  = (const float*)d_in[1]; // (V, T, L)
  const float* pvar = (const float*)d_in[2]; // (V, T, L)
  const float* ls   = (const float*)d_in[3]; // (L)
  const float* var  = (const float*)d_in[4]; // (L)

  float* out  = (float*)d_out;
  const size_t chains = (size_t)V_DIM * L_DIM; // 512
  float* m_fs = out;                               // chains*T*2
  float* P_fs = m_fs + chains * T_DIM * 2;         // chains*T*4
  float* mZ   = P_fs + chains * T_DIM * 4;         // chains*T
  float* PZ   = mZ + chains * T_DIM;               // chains*T
  float* slp  = PZ + chains * T_DIM;               // V

  const dim3 grid((unsigned)(chains / 32)); // 16 blocks of one wave32 each
  ssm_fwd_kernel<<<grid, 32, 0, stream>>>(dt, ypy, pvar, ls, var, m_fs, P_fs, slp);
  ssm_bwd_kernel<<<grid, 32, 0, stream>>>(dt, ls, var, m_fs, P_fs, mZ, PZ);
}